// GAT_76312978915565
// MI455X (gfx1250) — compile-verified
//
#include <hip/hip_runtime.h>
#include <hip/hip_bf16.h>

// ---------------------------------------------------------------------------
// GAT 2-layer pipeline for MI455X (gfx1250, wave32).
//  - Edge projections contracted analytically to [E,64]x[64,5] matvecs.
//  - Node projections via V_WMMA_F32_16X16X4_F32 (exact fp32 math).
//  - Edge softmax: monotonic-uint atomic max + atomicAdd f32 (L2-resident).
// ---------------------------------------------------------------------------

typedef float v2f __attribute__((ext_vector_type(2)));
typedef float v8f __attribute__((ext_vector_type(8)));

#define NEG_SLOPE 0.2f
// mapped value of -inf under the order-preserving float->uint transform
#define MAPPED_NEG_INF 0x007FFFFFu

__device__ __forceinline__ unsigned fmax_map(float f) {
    unsigned u = __float_as_uint(f);
    return (u & 0x80000000u) ? ~u : (u | 0x80000000u);
}
__device__ __forceinline__ float fmax_unmap(unsigned u) {
    return __uint_as_float((u & 0x80000000u) ? (u ^ 0x80000000u) : ~u);
}

// ---------------------------------------------------------------------------
// Kernel 1: fold attention vectors into edge-projection weights.
//   ve1[h*64+k] = sum_f ae1[h*32+f] * w_e1[(h*32+f)*64 + k]   (h<4, k<64)
//   ve2[k]      = sum_f ae2[f]      * w_e2[f*64 + k]          (k<64)
// ---------------------------------------------------------------------------
__global__ void prep_ve(const float* __restrict__ w_e1, const float* __restrict__ ae1,
                        const float* __restrict__ w_e2, const float* __restrict__ ae2,
                        float* __restrict__ ve1, float* __restrict__ ve2) {
    int t = threadIdx.x;
    if (t < 256) {
        int h = t >> 6, k = t & 63;
        float s = 0.f;
        for (int f = 0; f < 32; ++f) s += ae1[h * 32 + f] * w_e1[(h * 32 + f) * 64 + k];
        ve1[t] = s;
    } else if (t < 320) {
        int k = t - 256;
        float s = 0.f;
        for (int f = 0; f < 64; ++f) s += ae2[f] * w_e2[f * 64 + k];
        ve2[k] = s;
    }
}

// ---------------------------------------------------------------------------
// Kernel 2: WMMA fp32 GEMM  C[N,OC] = A[N,128] @ W[OC,128]^T
//  One block = 16 rows of A (LDS-staged); one wave per 16-col output tile.
//  K loop of 32 x V_WMMA_F32_16X16X4_F32.
//  A frag (16x4 f32): lane<16 -> row=lane, K=k+0/k+1; lane>=16 -> K=k+2/k+3.
//  B frag (4x16 f32): lane picks col o0+(lane&15), same K split.
//  D (16x16 f32): vgpr r, lane<16 -> M=r,N=lane; lane>=16 -> M=r+8.
// Requires N % 16 == 0 (N=50000 -> 3125 blocks) and K == 128.
// ---------------------------------------------------------------------------
__global__ void __launch_bounds__(256) wmma_gemm_xWT(
        const float* __restrict__ A, const float* __restrict__ W,
        float* __restrict__ C, int OC) {
    __shared__ float As[16 * 128];
    const int row0 = blockIdx.x * 16;

    // cooperative, fully-coalesced load of the contiguous 16x128 A tile
    const float4* Ag = (const float4*)(A + (size_t)row0 * 128);
    float4* Asv = (float4*)As;
    for (int i = threadIdx.x; i < 16 * 128 / 4; i += blockDim.x) Asv[i] = Ag[i];
    __syncthreads();

    const int lane = threadIdx.x & 31;
    const int wv   = threadIdx.x >> 5;
    const int o0   = wv * 16;
    const int mrow = lane & 15;
    const int half = lane >> 4;
    const int kofs = half * 2;

    const float* arow = As + mrow * 128;
    const float* wrow = W + (size_t)(o0 + mrow) * 128;

    v8f acc = {0.f, 0.f, 0.f, 0.f, 0.f, 0.f, 0.f, 0.f};
    #pragma unroll 4
    for (int k = 0; k < 128; k += 4) {
        v2f a = *(const v2f*)(arow + k + kofs);
        v2f b = *(const v2f*)(wrow + k + kofs);
        acc = __builtin_amdgcn_wmma_f32_16x16x4_f32(
            /*neg_a=*/false, a, /*neg_b=*/false, b,
            /*c_mod=*/(short)0, acc, /*reuse_a=*/false, /*reuse_b=*/false);
    }
    #pragma unroll
    for (int r = 0; r < 8; ++r)
        C[(size_t)(row0 + half * 8 + r) * OC + o0 + mrow] = acc[r];
}

// ---------------------------------------------------------------------------
// Kernel 3: per-node attention scores  eh[n,h] = feat[n,h,:].ah[h],  et likewise
// ---------------------------------------------------------------------------
__global__ void node_scores(const float* __restrict__ feat,
                            const float* __restrict__ ah, const float* __restrict__ at,
                            float* __restrict__ eh, float* __restrict__ et,
                            int N, int H, int F) {
    int t = blockIdx.x * blockDim.x + threadIdx.x;
    if (t >= N * H) return;
    int n = t / H, h = t - n * H;
    const float* fr = feat + (size_t)n * H * F + h * F;
    float a = 0.f, b = 0.f;
    for (int f = 0; f < F; ++f) {
        float v = fr[f];
        a += v * ah[h * F + f];
        b += v * at[h * F + f];
    }
    eh[t] = a;
    et[t] = b;
}

// ---------------------------------------------------------------------------
// Kernel 4: single streaming pass over edge_attr (410 MB) producing BOTH
//  layers' edge scores: ee1[E,4] and ee2[E].
//  Wave handles 2 edges: 32 lanes x float4 = 512 contiguous bytes per load.
//  Partial dots reduced with shfl_xor butterflies within 16-lane halves.
//  Requires E % 16 == 0 (E = 1.6M -> 100000 blocks of 256).
// ---------------------------------------------------------------------------
__global__ void edge_proj(const float* __restrict__ ea,
                          const float* __restrict__ ve1, const float* __restrict__ ve2,
                          float* __restrict__ ee1, float* __restrict__ ee2) {
    __shared__ float sv1[256];
    __shared__ float sv2[64];
    for (int i = threadIdx.x; i < 256; i += blockDim.x) sv1[i] = ve1[i];
    for (int i = threadIdx.x; i < 64; i += blockDim.x) sv2[i] = ve2[i];
    __syncthreads();

    const int lane = threadIdx.x & 31;
    const int wv   = threadIdx.x >> 5;
    const int sub  = lane & 15;
    const int half = lane >> 4;
    const long e = (long)blockIdx.x * 16 + wv * 2 + half;

    float4 v = *(const float4*)(ea + e * 64 + sub * 4);

    float acc[5];
    #pragma unroll
    for (int h = 0; h < 4; ++h) {
        float4 w = *(const float4*)(sv1 + h * 64 + sub * 4);
        acc[h] = v.x * w.x + v.y * w.y + v.z * w.z + v.w * w.w;
    }
    {
        float4 w = *(const float4*)(sv2 + sub * 4);
        acc[4] = v.x * w.x + v.y * w.y + v.z * w.z + v.w * w.w;
    }
    // butterfly reduce within each 16-lane half (xor masks 8,4,2,1 stay inside)
    #pragma unroll
    for (int off = 8; off > 0; off >>= 1) {
        #pragma unroll
        for (int j = 0; j < 5; ++j) acc[j] += __shfl_xor(acc[j], off, 32);
    }
    if (sub == 0) {
        ee1[e * 4 + 0] = acc[0];
        ee1[e * 4 + 1] = acc[1];
        ee1[e * 4 + 2] = acc[2];
        ee1[e * 4 + 3] = acc[3];
        ee2[e] = acc[4];
    }
}

// ---------------------------------------------------------------------------
// Init kernels (scratch & output are poisoned by the harness every capture).
// ---------------------------------------------------------------------------
__global__ void init_l1(float* __restrict__ agg, unsigned* __restrict__ m,
                        float* __restrict__ s, int N) {
    int t = blockIdx.x * blockDim.x + threadIdx.x;
    if (t < N * 128) agg[t] = 0.f;
    if (t < N * 4) { m[t] = MAPPED_NEG_INF; s[t] = 0.f; }
}

__global__ void init_l2(float* __restrict__ out, const float* __restrict__ b2,
                        unsigned* __restrict__ m, float* __restrict__ s, int N) {
    int t = blockIdx.x * blockDim.x + threadIdx.x;
    if (t < N * 64) out[t] = b2[t & 63];
    if (t < N) { m[t] = MAPPED_NEG_INF; s[t] = 0.f; }
}

// ---------------------------------------------------------------------------
// Edge softmax pass 1: raw score + leaky relu + segment max (atomic, mapped).
// ---------------------------------------------------------------------------
__global__ void edge_score(const int* __restrict__ src, const int* __restrict__ dst,
                           const float* __restrict__ eh, const float* __restrict__ et,
                           const float* __restrict__ ee, float* __restrict__ esc,
                           unsigned* __restrict__ m, int E, int H) {
    int t = blockIdx.x * blockDim.x + threadIdx.x;
    if (t >= E * H) return;
    int e = t / H, h = t - e * H;
    int sS = src[e], sD = dst[e];
    float v = eh[sS * H + h] + ee[t] + et[sD * H + h];
    v = (v > 0.f) ? v : NEG_SLOPE * v;
    esc[t] = v;
    atomicMax(&m[sD * H + h], fmax_map(v));
}

// ---------------------------------------------------------------------------
// Edge softmax pass 2: exp(e - max), segment sum (atomic add, L2 resident).
// ---------------------------------------------------------------------------
__global__ void edge_exp(const int* __restrict__ dst, const unsigned* __restrict__ m,
                         float* __restrict__ esc, float* __restrict__ s, int E, int H) {
    int t = blockIdx.x * blockDim.x + threadIdx.x;
    if (t >= E * H) return;
    int e = t / H, h = t - e * H;
    int d = dst[e];
    float ex = __expf(esc[t] - fmax_unmap(m[d * H + h]));
    esc[t] = ex;
    atomicAdd(&s[d * H + h], ex);
}

// ---------------------------------------------------------------------------
// Edge softmax pass 3: normalize + weighted scatter-aggregate.
//  feat gathers and agg atomics both hit the 192MB L2 (node arrays ~25MB).
// ---------------------------------------------------------------------------
__global__ void edge_agg(const int* __restrict__ src, const int* __restrict__ dst,
                         const float* __restrict__ feat, const float* __restrict__ esc,
                         const float* __restrict__ s, float* __restrict__ agg,
                         int total, int HF, int F, int H) {
    int t = blockIdx.x * blockDim.x + threadIdx.x;
    if (t >= total) return;
    int e = t / HF, f = t - e * HF;
    int h = f / F;
    int d = dst[e], sr = src[e];
    float a = esc[e * H + h] / s[d * H + h];
    atomicAdd(&agg[(size_t)d * HF + f], feat[(size_t)sr * HF + f] * a);
}

// ---------------------------------------------------------------------------
// Layer-1 epilogue: h1 = relu(agg + b1) (in place).
// ---------------------------------------------------------------------------
__global__ void relu_bias(float* __restrict__ agg, const float* __restrict__ b, int total) {
    int t = blockIdx.x * blockDim.x + threadIdx.x;
    if (t >= total) return;
    agg[t] = fmaxf(agg[t] + b[t & 127], 0.f);
}

// ---------------------------------------------------------------------------
extern "C" void kernel_launch(void* const* d_in, const int* in_sizes, int n_in,
                              void* d_out, int out_size, void* d_ws, size_t ws_size,
                              hipStream_t stream) {
    const float* x      = (const float*)d_in[0];
    const float* e_attr = (const float*)d_in[1];
    const int*   src    = (const int*)d_in[2];
    const int*   dst    = (const int*)d_in[3];
    const float* w_fc1  = (const float*)d_in[4];
    const float* w_e1   = (const float*)d_in[5];
    const float* ah1    = (const float*)d_in[6];
    const float* ae1    = (const float*)d_in[7];
    const float* at1    = (const float*)d_in[8];
    const float* b1     = (const float*)d_in[9];
    const float* w_fc2  = (const float*)d_in[10];
    const float* w_e2   = (const float*)d_in[11];
    const float* ah2    = (const float*)d_in[12];
    const float* ae2    = (const float*)d_in[13];
    const float* at2    = (const float*)d_in[14];
    const float* b2     = (const float*)d_in[15];
    float* out = (float*)d_out;

    const int N = in_sizes[0] / 128;   // 50000 (divisible by 16)
    const int E = in_sizes[2];         // 1600000 (divisible by 16)

    // workspace carve-up (~131 MB)
    float* ws = (float*)d_ws;
    size_t off = 0;
    auto take = [&](size_t n) { float* p = ws + off; off += n; return p; };
    float*    feat1 = take((size_t)N * 128);
    float*    agg1  = take((size_t)N * 128);   // becomes h1 after relu_bias
    float*    feat2 = take((size_t)N * 64);
    float*    eh1   = take((size_t)N * 4);
    float*    et1   = take((size_t)N * 4);
    float*    eh2   = take((size_t)N);
    float*    et2   = take((size_t)N);
    float*    ee1   = take((size_t)E * 4);
    float*    ee2   = take((size_t)E);
    float*    esc1  = take((size_t)E * 4);
    float*    esc2  = take((size_t)E);
    unsigned* m1    = (unsigned*)take((size_t)N * 4);
    float*    s1    = take((size_t)N * 4);
    unsigned* m2    = (unsigned*)take((size_t)N);
    float*    s2    = take((size_t)N);
    float*    ve1   = take(256);
    float*    ve2   = take(64);

    const int TB = 256;
    auto blocks = [&](long n) { return (int)((n + TB - 1) / TB); };

    // --- precompute folded edge-attention vectors -------------------------
    prep_ve<<<1, 320, 0, stream>>>(w_e1, ae1, w_e2, ae2, ve1, ve2);

    // --- layer 1 ----------------------------------------------------------
    wmma_gemm_xWT<<<N / 16, 256, 0, stream>>>(x, w_fc1, feat1, 128);
    node_scores<<<blocks((long)N * 4), TB, 0, stream>>>(feat1, ah1, at1, eh1, et1, N, 4, 32);
    edge_proj<<<E / 16, 256, 0, stream>>>(e_attr, ve1, ve2, ee1, ee2);
    init_l1<<<blocks((long)N * 128), TB, 0, stream>>>(agg1, m1, s1, N);
    edge_score<<<blocks((long)E * 4), TB, 0, stream>>>(src, dst, eh1, et1, ee1, esc1, m1, E, 4);
    edge_exp<<<blocks((long)E * 4), TB, 0, stream>>>(dst, m1, esc1, s1, E, 4);
    edge_agg<<<blocks((long)E * 128), TB, 0, stream>>>(src, dst, feat1, esc1, s1, agg1,
                                                       E * 128, 128, 32, 4);
    relu_bias<<<blocks((long)N * 128), TB, 0, stream>>>(agg1, b1, N * 128);

    // --- layer 2 ----------------------------------------------------------
    wmma_gemm_xWT<<<N / 16, 128, 0, stream>>>(agg1, w_fc2, feat2, 64);
    node_scores<<<blocks((long)N), TB, 0, stream>>>(feat2, ah2, at2, eh2, et2, N, 1, 64);
    init_l2<<<blocks((long)N * 64), TB, 0, stream>>>(out, b2, m2, s2, N);
    edge_score<<<blocks((long)E), TB, 0, stream>>>(src, dst, eh2, et2, ee2, esc2, m2, E, 1);
    edge_exp<<<blocks((long)E), TB, 0, stream>>>(dst, m2, esc2, s2, E, 1);
    edge_agg<<<blocks((long)E * 64), TB, 0, stream>>>(src, dst, feat2, esc2, s2, out,
                                                      E * 64, 64, 64, 1);
}